// GraphLayer_84447646974566
// MI455X (gfx1250) — compile-verified
//
#include <hip/hip_runtime.h>

// Problem constants (from reference setup_inputs): B=8, V=2048, C=256, fp32.
#define B_   8
#define V_   2048
#define C_   256
#define NROW (B_ * V_)          // 16384 rows of (B*V, C)
#define NT_  (V_ / 16)          // 128 u-tiles per batch

typedef __attribute__((ext_vector_type(2))) float v2f;
typedef __attribute__((ext_vector_type(8))) float v8f;

// V_WMMA_F32_16X16X4_F32: D(16x16,f32) = A(16x4,f32) * B(4x16,f32) + C
__device__ __forceinline__ v8f wmma_k4(v2f a, v2f b, v8f c) {
  return __builtin_amdgcn_wmma_f32_16x16x4_f32(false, a, false, b, (short)0, c,
                                               false, false);
}

// Async global->LDS copy (CDNA5 TDM-class path, tracked by ASYNCcnt).
// Per lane: copies 16 bytes from global address gaddr to LDS address lds_off, no VGPR round-trip.
// lds_off is the low 32 bits of the generic shared-space address (flat->LDS
// mapping truncates to addr[31:0], same value DS instructions use).
__device__ __forceinline__ void async_b128(unsigned lds_off,
                                           const float* gaddr) {
  asm volatile("global_load_async_to_lds_b128 %0, %1, off"
               :: "v"(lds_off), "v"(gaddr)
               : "memory");
}
#define WAIT_ASYNCCNT_0() asm volatile("s_wait_asynccnt 0x0" ::: "memory")

// ---------------------------------------------------------------------------
// sq[r] = sum_c x[r,c]^2   (one thread per row, float4 streaming)
// ---------------------------------------------------------------------------
__global__ void sq_kernel(const float* __restrict__ x, float* __restrict__ sq) {
  int r = blockIdx.x * blockDim.x + threadIdx.x;       // 0..NROW-1
  const float4* row = (const float4*)(x + (size_t)r * C_);
  float s = 0.f;
#pragma unroll 4
  for (int i = 0; i < C_ / 4; ++i) {
    float4 v = row[i];
    s += v.x * v.x + v.y * v.y + v.z * v.z + v.w * v.w;
  }
  sq[r] = s;
}

// ---------------------------------------------------------------------------
// h = x @ W^T  : h[r,o] = sum_k x[r,k] * W[o,k]
// One 16x16 output tile per wave; K=256 -> 64 fp32 WMMAs per tile.
// ---------------------------------------------------------------------------
__global__ __launch_bounds__(256) void linear_kernel(
    const float* __restrict__ x, const float* __restrict__ W,
    float* __restrict__ h) {
  const int lane = threadIdx.x & 31;
  const int wave = threadIdx.x >> 5;
  const int tile = blockIdx.x * 8 + wave;              // 0..16383
  const int rowTile = tile >> 4;                       // 1024 row tiles
  const int colTile = tile & 15;                       // 16 col tiles
  const int m  = lane & 15;
  const int hi = lane >> 4;                            // 0: K lo pair, 1: hi

  const float* xrow = x + (size_t)(rowTile * 16 + m) * C_;
  const float* wrow = W + (size_t)(colTile * 16 + m) * C_;

  v8f acc = {};
#pragma unroll 8
  for (int k0 = 0; k0 < C_; k0 += 4) {
    const int kk = k0 + hi * 2;
    v2f a = *(const v2f*)(xrow + kk);
    v2f b = *(const v2f*)(wrow + kk);
    acc = wmma_k4(a, b, acc);
  }
  const int mbase = rowTile * 16 + hi * 8;             // C/D: VGPR i -> M=i+hi*8
  const int col = colTile * 16 + m;
#pragma unroll
  for (int i = 0; i < 8; ++i)
    h[(size_t)(mbase + i) * C_ + col] = acc[i];
}

// ---------------------------------------------------------------------------
// Fused graph kernel with double-buffered async staging:
//   hp[v,:] = (1/rowsum) * sum_u sim(v,u) * h[u,:]
//   sim = 2/(exp(sqrt(max(|xv|^2+|xu|^2-2 xv.xu, 1e-12)))+1)
// Per u-tile: 8 waves split the K=256 Gram (8 WMMAs each), LDS-reduce,
// compute sim in-tile, accumulate L1 row sums, then each wave runs the
// aggregation for its 32 output columns (8 WMMAs). Next tile's x_u/h_u
// stream into the other LDS buffer via global_load_async_to_lds_b128,
// overlapping with the WMMA work; fenced by s_wait_asynccnt + barrier.
// ---------------------------------------------------------------------------
__global__ __launch_bounds__(256) void graph_kernel(
    const float* __restrict__ x, const float* __restrict__ h,
    const float* __restrict__ sq, float* __restrict__ hp) {
  __shared__ float s_xv[16 * C_];          // 16 KB
  __shared__ float s_xu[2][16 * C_];       // 32 KB (double buffered)
  __shared__ float s_hu[2][16 * C_];       // 32 KB (double buffered)
  __shared__ float s_gpart[8][256];        // per-wave Gram partials, 8 KB
  __shared__ float s_sim[256];             // 16x16 sim tile
  __shared__ float s_sqv[16];
  __shared__ float s_squ[16];
  __shared__ float s_rowsum[16];

  const int tid  = threadIdx.x;
  const int lane = tid & 31;
  const int wave = tid >> 5;
  const int m    = lane & 15;
  const int hi   = lane >> 4;
  const int batch = blockIdx.x >> 7;                   // NT_=128 tiles/batch
  const int vtile = blockIdx.x & (NT_ - 1);
  const size_t xbase = ((size_t)batch * V_ + vtile * 16) * C_;

  // Stage the 16x256 x_v tile once (regular loads; once per block).
  {
    const float4* src = (const float4*)(x + xbase);
    float4* dst = (float4*)s_xv;
    for (int i = tid; i < 16 * C_ / 4; i += 256) dst[i] = src[i];
  }
  if (tid < 16) {
    s_sqv[tid] = sq[batch * V_ + vtile * 16 + tid];
    s_rowsum[tid] = 0.f;
  }

  // Async stage of one u-tile (x_u + h_u, 32 KB) into LDS buffer `buf`.
  auto stage = [&](int ut, int buf) {
    const size_t ubase = ((size_t)batch * V_ + ut * 16) * C_;
    const float* gx = x + ubase;
    const float* gh = h + ubase;
#pragma unroll
    for (int j = 0; j < 4; ++j) {
      const int fo = (tid + 256 * j) * 4;              // float offset in tile
      async_b128((unsigned)(size_t)(&s_xu[buf][fo]), gx + fo);
      async_b128((unsigned)(size_t)(&s_hu[buf][fo]), gh + fo);
    }
  };

  v8f acc0 = {}, acc1 = {};            // this wave's 16 x 32 output columns
  const int col0 = wave * 32;

  stage(0, 0);                         // preflight prefetch

  for (int ut = 0; ut < NT_; ++ut) {
    const int cur = ut & 1;
    WAIT_ASYNCCNT_0();                 // my async writes for tile `ut` landed
    __syncthreads();                   // everyone's landed; prev iter done

    if (ut + 1 < NT_) stage(ut + 1, cur ^ 1);  // overlap next tile with WMMAs
    if (tid < 16) s_squ[tid] = sq[batch * V_ + ut * 16 + tid];

    // Gram partial: wave w covers K in [w*32, w*32+32)
    v8f gp = {};
#pragma unroll
    for (int j = 0; j < 8; ++j) {
      const int kk = wave * 32 + j * 4 + hi * 2;
      v2f a = *(const v2f*)(&s_xv[m * C_ + kk]);        // A[v, k]
      v2f b = *(const v2f*)(&s_xu[cur][m * C_ + kk]);   // B[k, u] = x_u[u, k]
      gp = wmma_k4(a, b, gp);
    }
#pragma unroll
    for (int i = 0; i < 8; ++i)
      s_gpart[wave][(i + hi * 8) * 16 + m] = gp[i];
    __syncthreads();

    // sim tile: reduce 8 wave partials, distance -> similarity
    {
      float g = 0.f;
#pragma unroll
      for (int w = 0; w < 8; ++w) g += s_gpart[w][tid];
      const int rM = tid >> 4, rN = tid & 15;
      float d = s_sqv[rM] + s_squ[rN] - 2.f * g;
      d = fmaxf(d, 1e-12f);
      s_sim[tid] = 2.f / (expf(sqrtf(d)) + 1.f);
    }
    __syncthreads();

    if (tid < 16) {                    // L1 row-sum accumulation (sim > 0)
      float rs = 0.f;
#pragma unroll
      for (int n = 0; n < 16; ++n) rs += s_sim[tid * 16 + n];
      s_rowsum[tid] += rs;
    }

    // Aggregation: acc(16x32) += sim(16x16) @ h_u(16x32), K=16 -> 4 steps
#pragma unroll
    for (int k0 = 0; k0 < 16; k0 += 4) {
      const int kk = k0 + hi * 2;
      v2f a;
      a.x = s_sim[m * 16 + kk];
      a.y = s_sim[m * 16 + kk + 1];
      v2f b0, b1;
      b0.x = s_hu[cur][kk * C_ + col0 + m];
      b0.y = s_hu[cur][(kk + 1) * C_ + col0 + m];
      b1.x = s_hu[cur][kk * C_ + col0 + 16 + m];
      b1.y = s_hu[cur][(kk + 1) * C_ + col0 + 16 + m];
      acc0 = wmma_k4(a, b0, acc0);
      acc1 = wmma_k4(a, b1, acc1);
    }
  }
  __syncthreads();

  // Normalize by L1 row sum and write hp (pre-BN).
#pragma unroll
  for (int i = 0; i < 8; ++i) {
    const int M = i + hi * 8;
    const float inv = 1.f / fmaxf(s_rowsum[M], 1e-12f);
    const size_t row = (size_t)batch * V_ + vtile * 16 + M;
    hp[row * C_ + col0 + m]      = acc0[i] * inv;
    hp[row * C_ + col0 + 16 + m] = acc1[i] * inv;
  }
}

// ---------------------------------------------------------------------------
// Per-channel batch statistics over (NROW, C): mean + rsqrt(var + eps)
// ---------------------------------------------------------------------------
__global__ __launch_bounds__(256) void stats_kernel(
    const float* __restrict__ hp, float* __restrict__ mean,
    float* __restrict__ rstd) {
  __shared__ float s1[256];
  __shared__ float s2[256];
  const int c = blockIdx.x;
  float a = 0.f, b = 0.f;
  for (int r = threadIdx.x; r < NROW; r += 256) {
    const float v = hp[(size_t)r * C_ + c];
    a += v;
    b += v * v;
  }
  s1[threadIdx.x] = a;
  s2[threadIdx.x] = b;
  __syncthreads();
  for (int s = 128; s > 0; s >>= 1) {
    if (threadIdx.x < s) {
      s1[threadIdx.x] += s1[threadIdx.x + s];
      s2[threadIdx.x] += s2[threadIdx.x + s];
    }
    __syncthreads();
  }
  if (threadIdx.x == 0) {
    const float mu = s1[0] / (float)NROW;
    const float var = s2[0] / (float)NROW - mu * mu;   // biased var (jnp.var)
    mean[c] = mu;
    rstd[c] = rsqrtf(var + 1e-5f);
  }
}

// ---------------------------------------------------------------------------
// out = 0.9*x + 0.1*LeakyReLU_0.1( (hp-mean)*rstd*bn_w + bn_b )
// ---------------------------------------------------------------------------
__global__ void final_kernel(const float* __restrict__ x,
                             const float* __restrict__ hp,
                             const float* __restrict__ mean,
                             const float* __restrict__ rstd,
                             const float* __restrict__ bn_w,
                             const float* __restrict__ bn_b,
                             float* __restrict__ out) {
  const int idx4 = blockIdx.x * blockDim.x + threadIdx.x;
  const size_t base = (size_t)idx4 * 4;
  const int c = (int)(base & (C_ - 1));
  const float4 xv = *(const float4*)(x + base);
  const float4 hv = *(const float4*)(hp + base);
  float xi[4] = {xv.x, xv.y, xv.z, xv.w};
  float hh[4] = {hv.x, hv.y, hv.z, hv.w};
  float o[4];
#pragma unroll
  for (int j = 0; j < 4; ++j) {
    float v = (hh[j] - mean[c + j]) * rstd[c + j] * bn_w[c + j] + bn_b[c + j];
    v = (v >= 0.f) ? v : 0.1f * v;
    o[j] = 0.9f * xi[j] + 0.1f * v;
  }
  *(float4*)(out + base) = make_float4(o[0], o[1], o[2], o[3]);
}

// ---------------------------------------------------------------------------
extern "C" void kernel_launch(void* const* d_in, const int* in_sizes, int n_in,
                              void* d_out, int out_size, void* d_ws,
                              size_t ws_size, hipStream_t stream) {
  const float* x    = (const float*)d_in[0];   // (B, V, C)
  const float* W    = (const float*)d_in[1];   // (C, C)
  const float* bn_w = (const float*)d_in[2];   // (C,)
  const float* bn_b = (const float*)d_in[3];   // (C,)
  float* out = (float*)d_out;

  // Workspace layout (floats): h | hp | sq | mean | rstd  (~33.6 MB)
  float* ws   = (float*)d_ws;
  float* h    = ws;                                  // NROW*C
  float* hp   = h + (size_t)NROW * C_;               // NROW*C
  float* sq   = hp + (size_t)NROW * C_;              // NROW
  float* mean = sq + NROW;                           // C
  float* rstd = mean + C_;                           // C

  sq_kernel<<<NROW / 256, 256, 0, stream>>>(x, sq);
  linear_kernel<<<(NROW / 16) * (C_ / 16) / 8, 256, 0, stream>>>(x, W, h);
  graph_kernel<<<B_ * NT_, 256, 0, stream>>>(x, h, sq, hp);
  stats_kernel<<<C_, 256, 0, stream>>>(hp, mean, rstd);
  final_kernel<<<(NROW * C_) / 1024, 256, 0, stream>>>(x, hp, mean, rstd, bn_w,
                                                       bn_b, out);
}